// GLMNB_60859686584630
// MI455X (gfx1250) — compile-verified
//
#include <hip/hip_runtime.h>
#include <math.h>

#define NVOX   300000
#define BDIM   256
#define NSTUDY 500

typedef float v2f __attribute__((ext_vector_type(2)));
typedef float v4f __attribute__((ext_vector_type(4)));
typedef float v8f __attribute__((ext_vector_type(8)));

#define MAIN_BLOCKS     586
#define THREADS         256
#define WAVES_PER_BLOCK (THREADS / 32)
#define TILES_PER_CHUNK 4
#define CHUNK_VOX       (TILES_PER_CHUNK * 16)                 // 64 voxels / wave pass
#define NCHUNKS         ((NVOX + CHUNK_VOX - 1) / CHUNK_VOX)   // 4688
#define NPAIRS          32   // 32 step-pairs * 2 WMMA * K=4 = 256 columns

// ws layout: float scalars [0..63] at offset 0
//   group g in floats f[g*16 + i]: i=0:r  i=1:A(=v*sum_mod)  i=2:B(=sum_mod2)
//                                  i=3..10: tt[y] for y=0..7
// double partials[MAIN_BLOCKS] at byte offset 512

// ---------------- setup: moderator sums, per-group scalars, log tables -------
__global__ void glmnb_setup(const float* __restrict__ mod1,
                            const float* __restrict__ mod2,
                            const float* __restrict__ gamma,
                            const float* __restrict__ asq1,
                            const float* __restrict__ asq2,
                            float* __restrict__ wsf) {
  __shared__ float s1[512], s1q[512], s2[512], s2q[512];
  int t = threadIdx.x;
  float m1 = 0.f, m2 = 0.f;
  float g0 = gamma[0], g1 = gamma[1], g2 = gamma[2], g3 = gamma[3];
  if (t < NSTUDY) {
    const float* r1 = mod1 + t * 4;
    const float* r2 = mod2 + t * 4;
    m1 = expf(r1[0] * g0 + r1[1] * g1 + r1[2] * g2 + r1[3] * g3);
    m2 = expf(r2[0] * g0 + r2[1] * g1 + r2[2] * g2 + r2[3] * g3);
  }
  s1[t] = m1; s1q[t] = m1 * m1; s2[t] = m2; s2q[t] = m2 * m2;
  __syncthreads();
  for (int off = 256; off > 0; off >>= 1) {
    if (t < off) {
      s1[t]  += s1[t + off];  s1q[t] += s1q[t + off];
      s2[t]  += s2[t + off];  s2q[t] += s2q[t + off];
    }
    __syncthreads();
  }
  if (t == 0) {
    const float fact[8] = {1.f, 1.f, 2.f, 6.f, 24.f, 120.f, 720.f, 5040.f};
    for (int g = 0; g < 2; ++g) {
      float sm    = g ? s2[0]  : s1[0];
      float sm2   = g ? s2q[0] : s1q[0];
      float as    = g ? asq2[0] : asq1[0];
      float alpha = as * as;
      float v     = 1.0f / alpha;
      float r     = v * sm * sm / sm2;   // scalar per group
      float A     = v * sm;              // p = mu*B / (A + mu*B)
      float B     = sm2;
      float* f = wsf + g * 16;
      f[0] = r; f[1] = A; f[2] = B;
      // tt[y] = sum_{k<y} log(r+k) - log(y!)  (== lgamma(y+r)-lgamma(y+1)-lgamma(r))
      float pre = 0.f;
      for (int y = 0; y < 8; ++y) {
        f[3 + y] = (y == 0) ? 0.f : (pre - logf(fact[y]));
        pre += logf(r + (float)y);
      }
    }
  }
}

// ---------------- main: WMMA mat-vec (both groups) + NB log-likelihood -------
__global__ void __launch_bounds__(THREADS)
glmnb_main(const float* __restrict__ Coef,
           const float* __restrict__ beta1,
           const float* __restrict__ beta2,
           const int*   __restrict__ foci1,
           const int*   __restrict__ foci2,
           const float* __restrict__ wsf,
           double*      __restrict__ partials) {
  // A-operand table, one v4f per (step-pair, lane) with the half-split K map:
  // half0 lanes cover cols [4p..4p+3], half1 lanes cover cols [128+4p..128+4p+3].
  __shared__ __align__(16) v4f sATab[NPAIRS * 32];   // 16 KB
  __shared__ float  sF[32];
  __shared__ double sPart[WAVES_PER_BLOCK];

  const int tid = threadIdx.x;

  for (int e = tid; e < NPAIRS * 32; e += THREADS) {
    int l  = e & 31;                         // lane this entry feeds
    int p  = e >> 5;                         // step-pair
    int me = l & 15;                         // A-matrix row (M)
    int kc = ((l >> 4) << 7) + (p << 2);     // column base: half*128 + 4p
    v4f a = {0.f, 0.f, 0.f, 0.f};
    if (me == 0)      a = (v4f){beta1[kc], beta1[kc+1], beta1[kc+2], beta1[kc+3]};
    else if (me == 1) a = (v4f){beta2[kc], beta2[kc+1], beta2[kc+2], beta2[kc+3]};
    sATab[e] = a;
  }
  if (tid < 32) sF[tid] = wsf[tid];
  __syncthreads();

  const float r1 = sF[0],  A1 = sF[1],  B1 = sF[2];
  const float r2 = sF[16], A2 = sF[17], B2 = sF[18];

  const int lane        = tid & 31;
  const int waveInBlock = tid >> 5;
  const int wave        = blockIdx.x * WAVES_PER_BLOCK + waveInBlock;
  const int nWaves      = gridDim.x * WAVES_PER_BLOCK;

  const int      m     = lane & 15;                    // B-column (voxel) in tile
  const unsigned halfB = (unsigned)((lane >> 4) << 9); // half1 reads cols 128.. (+512B)

  double lacc = 0.0;

  for (int chunk = wave; chunk < NCHUNKS; chunk += nWaves) {
    const int base = chunk * CHUNK_VOX;

    unsigned vOff[TILES_PER_CHUNK];
#pragma unroll
    for (int t = 0; t < TILES_PER_CHUNK; ++t) {
      int row = base + t * 16 + m;
      row = (row < NVOX) ? row : (NVOX - 1);           // clamp; masked later
      vOff[t] = (unsigned)row * (unsigned)(BDIM * 4) + halfB;
    }

    v8f c[TILES_PER_CHUNK] = {};

    auto loadPair = [&](int p, v4f& a, v4f* b) {
      a = sATab[p * 32 + lane];                        // ds_load_b128
#pragma unroll
      for (int t = 0; t < TILES_PER_CHUNK; ++t)        // global_load_b128 (saddr)
        b[t] = *(const v4f*)((const char*)Coef +
                             (size_t)(vOff[t] + ((unsigned)p << 4)));
    };
    auto mm = [&](const v4f& a, const v4f* b) {
      v2f a0 = {a.x, a.y}, a1 = {a.z, a.w};
#pragma unroll
      for (int t = 0; t < TILES_PER_CHUNK; ++t) {
        v2f bl = {b[t].x, b[t].y};
        c[t] = __builtin_amdgcn_wmma_f32_16x16x4_f32(
            false, a0, false, bl, (short)0, c[t], false, false);
      }
#pragma unroll
      for (int t = 0; t < TILES_PER_CHUNK; ++t) {
        v2f bh = {b[t].z, b[t].w};
        c[t] = __builtin_amdgcn_wmma_f32_16x16x4_f32(
            false, a1, false, bh, (short)0, c[t], false, false);
      }
    };

    // depth-2 software pipeline: loads for pair p+2 issue while pairs p, p+1
    // are still unconsumed -> ~10+ outstanding loads/wave, no wait-0 chains.
    // unroll 3 == rotation period of the 3 buffer sets -> pure register
    // renaming, no v_mov rotation copies (30 iterations / 3 == 10 bodies).
    v4f aA, aB, aN;
    v4f bA[TILES_PER_CHUNK], bB[TILES_PER_CHUNK], bN[TILES_PER_CHUNK];
    loadPair(0, aA, bA);
    loadPair(1, aB, bB);
#pragma unroll 3
    for (int p = 0; p + 2 < NPAIRS; ++p) {             // p = 0..29
      loadPair(p + 2, aN, bN);
      mm(aA, bA);
      aA = aB; aB = aN;
#pragma unroll
      for (int t = 0; t < TILES_PER_CHUNK; ++t) { bA[t] = bB[t]; bB[t] = bN[t]; }
    }
    mm(aA, bA);
    mm(aB, bB);

    // lanes 0..15 hold D[M=0,n]=dot_g1 (c[.][0]) and D[M=1,n]=dot_g2 (c[.][1])
    if (lane < 16) {
#pragma unroll
      for (int t = 0; t < TILES_PER_CHUNK; ++t) {
        int vox = base + t * 16 + lane;
        if (vox < NVOX) {
          // group 1
          float mu = expf(c[t][0]);
          float z  = mu * B1;
          float pp = z / (A1 + z);
          int   y  = foci1[vox];
          y = (y < 0) ? 0 : ((y > 7) ? 7 : y);
          lacc += (double)(sF[3 + y] + r1 * log1pf(-pp) + (float)y * logf(pp));
          // group 2
          mu = expf(c[t][1]);
          z  = mu * B2;
          pp = z / (A2 + z);
          y  = foci2[vox];
          y = (y < 0) ? 0 : ((y > 7) ? 7 : y);
          lacc += (double)(sF[19 + y] + r2 * log1pf(-pp) + (float)y * logf(pp));
        }
      }
    }
  }

  // deterministic wave reduction, then fixed-order block sum
  for (int off = 16; off > 0; off >>= 1)
    lacc += __shfl_down(lacc, off, 32);
  if (lane == 0) sPart[waveInBlock] = lacc;
  __syncthreads();
  if (tid == 0) {
    double s = 0.0;
    for (int w = 0; w < WAVES_PER_BLOCK; ++w) s += sPart[w];
    partials[blockIdx.x] = s;
  }
}

// ---------------- finalize: deterministic sum of block partials --------------
__global__ void glmnb_finalize(const double* __restrict__ partials, int n,
                               float* __restrict__ out) {
  __shared__ double sd[256];
  int t = threadIdx.x;
  double s = 0.0;
  for (int i = t; i < n; i += 256) s += partials[i];
  sd[t] = s;
  __syncthreads();
  for (int off = 128; off > 0; off >>= 1) {
    if (t < off) sd[t] += sd[t + off];
    __syncthreads();
  }
  if (t == 0) out[0] = (float)(-sd[0]);
}

extern "C" void kernel_launch(void* const* d_in, const int* in_sizes, int n_in,
                              void* d_out, int out_size, void* d_ws, size_t ws_size,
                              hipStream_t stream) {
  const float* Coef  = (const float*)d_in[0];
  const float* mod1  = (const float*)d_in[1];
  const float* mod2  = (const float*)d_in[2];
  const float* beta1 = (const float*)d_in[3];
  const float* beta2 = (const float*)d_in[4];
  const float* asq1  = (const float*)d_in[5];
  const float* asq2  = (const float*)d_in[6];
  const float* gamma = (const float*)d_in[7];
  const int*   foci1 = (const int*)d_in[8];
  const int*   foci2 = (const int*)d_in[9];
  // d_in[10], d_in[11] (foci_per_study) unused by the reference math

  float*  wsf      = (float*)d_ws;
  double* partials = (double*)((char*)d_ws + 512);
  float*  out      = (float*)d_out;

  glmnb_setup<<<1, 512, 0, stream>>>(mod1, mod2, gamma, asq1, asq2, wsf);
  glmnb_main<<<MAIN_BLOCKS, THREADS, 0, stream>>>(Coef, beta1, beta2,
                                                  foci1, foci2, wsf, partials);
  glmnb_finalize<<<1, 256, 0, stream>>>(partials, MAIN_BLOCKS, out);
}